// Model_89386859364699
// MI455X (gfx1250) — compile-verified
//
#include <hip/hip_runtime.h>

typedef __attribute__((ext_vector_type(2))) float v2f;
typedef __attribute__((ext_vector_type(8))) float v8f;

#define NEG_SLOPE 0.2f

// ---- monotonic float<->uint encoding for atomic max on floats ----
__device__ __forceinline__ unsigned fenc(float f) {
    unsigned u = __float_as_uint(f);
    return (u & 0x80000000u) ? ~u : (u | 0x80000000u);
}
__device__ __forceinline__ float fdec(unsigned e) {
    return (e & 0x80000000u) ? __uint_as_float(e & 0x7FFFFFFFu)
                             : __uint_as_float(~e);
}
#define ENC_NEG_INF 0x007FFFFFu  // fenc(-inf)

// ---------------------------------------------------------------------------
// Pack W[K][dout] into transposed, zero-padded Wt[dout_pad][K] so the WMMA
// B-fragment {W[k][n], W[k+1][n]} is a contiguous, unguarded float2 load.
// ---------------------------------------------------------------------------
__global__ void pack_wt(const float* __restrict__ W, float* __restrict__ Wt,
                        int K, int dout, int dout_pad) {
    int t = blockIdx.x * blockDim.x + threadIdx.x;
    if (t >= dout_pad * K) return;
    int n = t / K;
    int k = t - n * K;
    Wt[t] = (n < dout) ? W[(size_t)k * dout + n] : 0.0f;
}

// ---------------------------------------------------------------------------
// H = X @ W via V_WMMA_F32_16X16X4_F32.
//  - Block = 256 threads (8 waves). All waves share one 64-column B tile,
//    staged once into LDS (contiguous slab of Wt), read via ds_load_b64.
//  - Each wave owns a 16x64 output tile (4 accumulators); per k-step one
//    prefetched A float2 feeds 4 back-to-back WMMAs.
//  - A fragment is software-pipelined one iteration ahead so WMMAs never
//    wait on a just-issued global load.
// ---------------------------------------------------------------------------
#define FC_LDS_MAXW 122  // max (din + 2): 120 + 2 pad floats (bank spread)

__global__ void __launch_bounds__(256)
gat_fc_wmma(const float* __restrict__ X,
            const float* __restrict__ Wt,   // [dout_pad][din]
            float* __restrict__ H,
            int N, int din, int dout) {
    __shared__ float blds[64 * FC_LDS_MAXW];

    const int wave = threadIdx.x >> 5;
    const int lane = threadIdx.x & 31;
    const int n0   = blockIdx.y * 64;
    const int ldsw = din + 2;               // padded LDS row stride (even)

    // Cooperative fill: Wt rows [n0, n0+64) are one contiguous slab.
    {
        const float* __restrict__ src = Wt + (size_t)n0 * din;
        for (int n = wave * 8; n < wave * 8 + 8; ++n) {
            const float* __restrict__ s = src + (size_t)n * din;
            float* __restrict__ q = &blds[n * ldsw];
            for (int k = lane; k < din; k += 32) q[k] = s[k];
        }
    }
    __syncthreads();

    const int Mtiles = (N + 15) >> 4;
    const int mt     = blockIdx.x * 8 + wave;
    if (mt >= Mtiles) return;               // uniform per-wave exit
    const int m0 = mt * 16;

    int m = m0 + (lane & 15);
    if (m >= N) m = N - 1;                  // clamp loads; stores guarded
    const int koff = (lane >> 4) * 2;       // upper half-wave: K+2,K+3

    const float* __restrict__ xrow = X + (size_t)m * din + koff;
    const float* b0p = &blds[(lane & 15) * ldsw + koff];
    const float* b1p = b0p + 16 * ldsw;
    const float* b2p = b0p + 32 * ldsw;
    const float* b3p = b0p + 48 * ldsw;

    v8f acc[4] = {v8f{}, v8f{}, v8f{}, v8f{}};
    v2f a = *(const v2f*)(xrow);            // pipelined A fragment
    int k0 = 0;
    for (; k0 < din - 4; k0 += 4) {
        v2f an = *(const v2f*)(xrow + k0 + 4);      // prefetch next A
        v2f b0 = *(const v2f*)(b0p + k0);
        v2f b1 = *(const v2f*)(b1p + k0);
        v2f b2 = *(const v2f*)(b2p + k0);
        v2f b3 = *(const v2f*)(b3p + k0);
        acc[0] = __builtin_amdgcn_wmma_f32_16x16x4_f32(
                     false, a, false, b0, (short)0, acc[0], false, false);
        acc[1] = __builtin_amdgcn_wmma_f32_16x16x4_f32(
                     false, a, false, b1, (short)0, acc[1], false, false);
        acc[2] = __builtin_amdgcn_wmma_f32_16x16x4_f32(
                     false, a, false, b2, (short)0, acc[2], false, false);
        acc[3] = __builtin_amdgcn_wmma_f32_16x16x4_f32(
                     false, a, false, b3, (short)0, acc[3], false, false);
        a = an;
    }
    {   // peeled last k-step (no A prefetch past the row end)
        v2f b0 = *(const v2f*)(b0p + k0);
        v2f b1 = *(const v2f*)(b1p + k0);
        v2f b2 = *(const v2f*)(b2p + k0);
        v2f b3 = *(const v2f*)(b3p + k0);
        acc[0] = __builtin_amdgcn_wmma_f32_16x16x4_f32(
                     false, a, false, b0, (short)0, acc[0], false, false);
        acc[1] = __builtin_amdgcn_wmma_f32_16x16x4_f32(
                     false, a, false, b1, (short)0, acc[1], false, false);
        acc[2] = __builtin_amdgcn_wmma_f32_16x16x4_f32(
                     false, a, false, b2, (short)0, acc[2], false, false);
        acc[3] = __builtin_amdgcn_wmma_f32_16x16x4_f32(
                     false, a, false, b3, (short)0, acc[3], false, false);
    }

    // D layout: VGPR r -> lanes 0-15: (M=r, N=lane); lanes 16-31: (M=8+r).
    const int rbase = m0 + ((lane >> 4) << 3);
#pragma unroll
    for (int t = 0; t < 4; ++t) {
        const int col = n0 + t * 16 + (lane & 15);
        if (col < dout) {
#pragma unroll
            for (int r = 0; r < 8; ++r) {
                const int row = rbase + r;
                if (row < N) H[(size_t)row * dout + col] = acc[t][r];
            }
        }
    }
}

// ---------------------------------------------------------------------------
// el[i] = h[i,:] . al ; er[i] = h[i,:] . ar
// ---------------------------------------------------------------------------
__global__ void node_dots(const float* __restrict__ H,
                          const float* __restrict__ al,
                          const float* __restrict__ ar,
                          float* __restrict__ el, float* __restrict__ er,
                          int N, int dout) {
    int i = blockIdx.x * blockDim.x + threadIdx.x;
    if (i >= N) return;
    const float* __restrict__ h = H + (size_t)i * dout;
    float sl = 0.0f, sr = 0.0f;
    for (int j = 0; j < dout; ++j) {
        float v = h[j];
        sl += v * al[j];
        sr += v * ar[j];
    }
    el[i] = sl;
    er[i] = sr;
}

// ---------------------------------------------------------------------------
// Per-layer init: m = -inf (encoded), denom = 0, out[i,:] = bias (the edge
// scatter then accumulates on top, matching segment_sum(...) + b).
// ---------------------------------------------------------------------------
__global__ void layer_init(unsigned* __restrict__ m_enc,
                           float* __restrict__ denom,
                           float* __restrict__ out,
                           const float* __restrict__ b,
                           int N, int dout) {
    int i = blockIdx.x * blockDim.x + threadIdx.x;
    if (i >= N) return;
    m_enc[i] = ENC_NEG_INF;
    denom[i] = 0.0f;
    float* o = out + (size_t)i * dout;
    for (int j = 0; j < dout; ++j) o[j] = b[j];
}

// ---------------------------------------------------------------------------
// Pass 1 over edges: e = leaky_relu(el[src]+er[dst]); segment max over dst.
// ---------------------------------------------------------------------------
__global__ void edge_score_max(const int* __restrict__ ei,
                               const float* __restrict__ el,
                               const float* __restrict__ er,
                               float* __restrict__ ebuf,
                               unsigned* __restrict__ m_enc, int E) {
    int t = blockIdx.x * blockDim.x + threadIdx.x;
    if (t >= E) return;
    int s = ei[t];
    int d = ei[E + t];
    float e = el[s] + er[d];
    e = (e > 0.0f) ? e : NEG_SLOPE * e;
    ebuf[t] = e;
    atomicMax(m_enc + d, fenc(e));
}

// ---------------------------------------------------------------------------
// Pass 2: ex = exp(e - m[dst]); denom[dst] += ex.
// ---------------------------------------------------------------------------
__global__ void edge_exp_sum(const int* __restrict__ ei,
                             const unsigned* __restrict__ m_enc,
                             float* __restrict__ ebuf,
                             float* __restrict__ denom, int E) {
    int t = blockIdx.x * blockDim.x + threadIdx.x;
    if (t >= E) return;
    int d = ei[E + t];
    float ex = __expf(ebuf[t] - fdec(m_enc[d]));
    ebuf[t] = ex;
    atomicAdd(denom + d, ex);
}

// ---------------------------------------------------------------------------
// Pass 3: one wave32 per edge; lanes sweep the feature dim so h[src] loads
// and out[dst] atomics are coalesced. h/out stay L2-resident (<=48MB each).
// ---------------------------------------------------------------------------
__global__ void edge_scatter(const int* __restrict__ ei,
                             const float* __restrict__ ebuf,
                             const float* __restrict__ denom,
                             const float* __restrict__ H,
                             float* __restrict__ out, int E, int dout) {
    int e = (blockIdx.x * blockDim.x + threadIdx.x) >> 5;
    int lane = threadIdx.x & 31;
    if (e >= E) return;
    int s = ei[e];
    int d = ei[E + e];
    float alpha = ebuf[e] / (denom[d] + 1e-9f);
    const float* __restrict__ h = H + (size_t)s * dout;
    float* o = out + (size_t)d * dout;
    for (int j = lane; j < dout; j += 32)
        atomicAdd(o + j, h[j] * alpha);
}

// ---------------------------------------------------------------------------
// Host-side driver for one GAT layer.
// ---------------------------------------------------------------------------
static void run_layer(const float* x, int din, int dout,
                      const int* ei, int E, int N,
                      const float* W, const float* al, const float* ar,
                      const float* b, float* out,
                      float* h_buf, float* wt_buf, float* el, float* er,
                      unsigned* m_enc, float* denom, float* ebuf,
                      hipStream_t stream) {
    const int dout_pad = ((dout + 63) / 64) * 64;   // 64 or 128

    const int pk = dout_pad * din;
    pack_wt<<<(pk + 255) / 256, 256, 0, stream>>>(W, wt_buf, din, dout, dout_pad);

    const int Mtiles = (N + 15) / 16;
    dim3 gg((Mtiles + 7) / 8, dout_pad / 64);
    gat_fc_wmma<<<gg, 256, 0, stream>>>(x, wt_buf, h_buf, N, din, dout);

    const int nb = (N + 255) / 256;
    node_dots<<<nb, 256, 0, stream>>>(h_buf, al, ar, el, er, N, dout);
    layer_init<<<nb, 256, 0, stream>>>(m_enc, denom, out, b, N, dout);

    const int eb = (E + 255) / 256;
    edge_score_max<<<eb, 256, 0, stream>>>(ei, el, er, ebuf, m_enc, E);
    edge_exp_sum<<<eb, 256, 0, stream>>>(ei, m_enc, ebuf, denom, E);

    const int sb = (E + 7) / 8;  // 8 edges (waves) per 256-thread block
    edge_scatter<<<sb, 256, 0, stream>>>(ei, ebuf, denom, h_buf, out, E, dout);
}

extern "C" void kernel_launch(void* const* d_in, const int* in_sizes, int n_in,
                              void* d_out, int out_size, void* d_ws, size_t ws_size,
                              hipStream_t stream) {
    const float* x   = (const float*)d_in[0];
    const int*   ei0 = (const int*)d_in[1];
    const int*   ei1 = (const int*)d_in[2];
    const int N = in_sizes[0] / 120;
    const int E = in_sizes[1] / 2;

    const float* W1  = (const float*)d_in[3];
    const float* al1 = (const float*)d_in[4];
    const float* ar1 = (const float*)d_in[5];
    const float* b1  = (const float*)d_in[6];
    const float* W2  = (const float*)d_in[7];
    const float* al2 = (const float*)d_in[8];
    const float* ar2 = (const float*)d_in[9];
    const float* b2  = (const float*)d_in[10];
    const float* W3  = (const float*)d_in[11];
    const float* al3 = (const float*)d_in[12];
    const float* ar3 = (const float*)d_in[13];
    const float* b3  = (const float*)d_in[14];
    const float* W4  = (const float*)d_in[15];
    const float* al4 = (const float*)d_in[16];
    const float* ar4 = (const float*)d_in[17];
    const float* b4  = (const float*)d_in[18];

    // Workspace layout (floats)
    float*    ws     = (float*)d_ws;
    float*    h_buf  = ws;                             // N*120
    float*    o_mid  = h_buf + (size_t)N * 120;        // N*60 (L1 out, L3 out)
    float*    el     = o_mid + (size_t)N * 60;         // N
    float*    er     = el + N;                         // N
    unsigned* m_enc  = (unsigned*)(er + N);            // N
    float*    denom  = (float*)(m_enc + N);            // N
    float*    ebuf   = denom + N;                      // E
    float*    wt_buf = ebuf + E;                       // 128*120 max

    float* out  = (float*)d_out;
    float* outA = out;                                 // h_a : N*120
    float* outF = out + (size_t)N * 120;               // h_f : N*120

    // h1 = GAT(x, ei0; W1);  h_a = GAT(h1, ei1; W2)
    run_layer(x,     120, 60,  ei0, E, N, W1, al1, ar1, b1, o_mid,
              h_buf, wt_buf, el, er, m_enc, denom, ebuf, stream);
    run_layer(o_mid,  60, 120, ei1, E, N, W2, al2, ar2, b2, outA,
              h_buf, wt_buf, el, er, m_enc, denom, ebuf, stream);

    // h2 = GAT(x, ei0; W3);  h_f = GAT(h2, ei1; W4)
    run_layer(x,     120, 60,  ei0, E, N, W3, al3, ar3, b3, o_mid,
              h_buf, wt_buf, el, er, m_enc, denom, ebuf, stream);
    run_layer(o_mid,  60, 120, ei1, E, N, W4, al4, ar4, b4, outF,
              h_buf, wt_buf, el, er, m_enc, denom, ebuf, stream);
}